// RobertaSelfAttention_matchKV_71502615544172
// MI455X (gfx1250) — compile-verified
//
#include <hip/hip_runtime.h>
#include <hip/hip_bf16.h>

typedef __attribute__((ext_vector_type(2))) float v2f;
typedef __attribute__((ext_vector_type(8))) float v8f;
typedef __attribute__((ext_vector_type(4))) float f4;

#define BSZ 8
#define LEN 2048
#define HID 1024
#define NHEAD 16
#define HD 64
#define MM (BSZ * LEN)      // 16384 rows
#define NN (NHEAD * HD)     // 1024 cols
#define KDIM HID            // 1024 reduction

// GEMM tiling
#define BM 64
#define BN 128
#define BK 16
#define LDA (BK + 4)        // padded LDS stride for A (80 B rows, 16B aligned)
#define LDB (BN + 4)        // padded LDS stride for B (528 B rows, 16B aligned)
#define NTILES (KDIM / BK)  // 64 k-tiles

// --------------------------------------------------------------------------
// CDNA5 async copy helpers: GLOBAL_LOAD_ASYNC_TO_LDS_B128 (ASYNCcnt-tracked).
// LDS byte offset = low 32 bits of the shared-aperture flat address
// (ISA 10.2).
// --------------------------------------------------------------------------
__device__ __forceinline__ uint32_t lds_off(const void* p) {
    return (uint32_t)(uintptr_t)p;
}
__device__ __forceinline__ void async_copy_b128(uint32_t lds, const void* gptr) {
    asm volatile("global_load_async_to_lds_b128 %0, %1, off"
                 :: "v"(lds), "v"(gptr) : "memory");
}
__device__ __forceinline__ void wait_async0() {
    asm volatile("s_wait_asynccnt 0" ::: "memory");
}

// ---------------------------------------------------------------------------
// Fused GEMM: T = relu(X @ W + bias).
//   WITH_DOTS=0 (V path):  store T to Out.
//   WITH_DOTS=1 (K path):  never materialize T; accumulate per-(row,head)
//                          partials of T . RH into global dots[] via atomics.
// 8 waves / block; block tile 64x128; wave = 32x32 sub-tile = 2x2 WMMA accs.
// Double-buffered LDS fed by global_load_async_to_lds_b128.
// ---------------------------------------------------------------------------
template <bool WITH_DOTS>
__global__ __launch_bounds__(256)
void gemm_fused_kernel(const float* __restrict__ X,
                       const float* __restrict__ W,
                       const float* __restrict__ bias,
                       const float* __restrict__ RH,   // [NN] flat (head-major)
                       float* __restrict__ Out,        // V path
                       float* __restrict__ dots)       // K path
{
    __shared__ __align__(16) float As[2][BM][LDA];
    __shared__ __align__(16) float Bs[2][BK][LDB];

    const int mBlock = blockIdx.x * BM;
    const int nBlock = blockIdx.y * BN;
    const int tid  = threadIdx.x;
    const int lane = tid & 31;
    const int w    = tid >> 5;          // wave 0..7
    const int wm   = (w & 1) * 32;      // wave row offset in block tile
    const int wn   = (w >> 1) * 32;     // wave col offset in block tile
    const int lh   = lane >> 4;         // half-wave select
    const int lm   = lane & 15;

    v8f acc[2][2] = {};

    // Global->LDS staging assignments (one b128 for A, two for B, per lane)
    const int aRow = tid >> 2;          // 0..63
    const int aCol = (tid & 3) * 4;     // 0,4,8,12
    const int bRow = tid >> 5;          // 0..7 (also +8)
    const int bCol = (tid & 31) * 4;    // 0..124

    const float* gA  = X + (size_t)(mBlock + aRow) * KDIM + aCol;
    const float* gB0 = W + (size_t)bRow       * NN + nBlock + bCol;
    const float* gB1 = W + (size_t)(bRow + 8) * NN + nBlock + bCol;

    const uint32_t lA[2]  = { lds_off(&As[0][aRow][aCol]),     lds_off(&As[1][aRow][aCol]) };
    const uint32_t lB0[2] = { lds_off(&Bs[0][bRow][bCol]),     lds_off(&Bs[1][bRow][bCol]) };
    const uint32_t lB1[2] = { lds_off(&Bs[0][bRow + 8][bCol]), lds_off(&Bs[1][bRow + 8][bCol]) };

    // Prime the pipeline: tile 0 -> buffer 0
    async_copy_b128(lA[0],  gA);
    async_copy_b128(lB0[0], gB0);
    async_copy_b128(lB1[0], gB1);

    #pragma unroll 2
    for (int t = 0; t < NTILES; ++t) {
        const int cur = t & 1;
        wait_async0();          // my async copies for tile t have landed
        __syncthreads();        // everyone's landed; previous buffer now free

        if (t + 1 < NTILES) {   // overlap tile t+1 copy with tile t compute
            const int nk = (t + 1) * BK;
            async_copy_b128(lA[cur ^ 1],  gA + nk);
            async_copy_b128(lB0[cur ^ 1], gB0 + (size_t)nk * NN);
            async_copy_b128(lB1[cur ^ 1], gB1 + (size_t)nk * NN);
        }

        #pragma unroll
        for (int kk = 0; kk < BK; kk += 4) {
            v2f a[2], b[2];
            #pragma unroll
            for (int i = 0; i < 2; ++i) {
                // A 16x4 frag: lanes 0-15 K={kk,kk+1}, lanes 16-31 K={kk+2,kk+3}
                const float* p = &As[cur][wm + i * 16 + lm][kk + lh * 2];
                a[i].x = p[0];
                a[i].y = p[1];
            }
            #pragma unroll
            for (int j = 0; j < 2; ++j) {
                const int col = wn + j * 16 + lm;
                b[j].x = Bs[cur][kk + lh * 2][col];
                b[j].y = Bs[cur][kk + lh * 2 + 1][col];
            }
            #pragma unroll
            for (int i = 0; i < 2; ++i)
                #pragma unroll
                for (int j = 0; j < 2; ++j)
                    acc[i][j] = __builtin_amdgcn_wmma_f32_16x16x4_f32(
                        false, a[i], false, b[j], (short)0, acc[i][j],
                        false, false);
        }
    }

    // ---- Epilogue ----
    float bv[2], rhv[2];
    #pragma unroll
    for (int j = 0; j < 2; ++j) {
        const int col = nBlock + wn + j * 16 + lm;
        bv[j] = bias[col];
        if (WITH_DOTS) rhv[j] = RH[col];
    }

    if (WITH_DOTS) {
        // All 4 tiles of a wave lie in one head (wn spans 32 cols, head = 64).
        const int head = (nBlock + wn) >> 6;
        #pragma unroll
        for (int i = 0; i < 2; ++i) {
            #pragma unroll
            for (int r = 0; r < 8; ++r) {
                float s = 0.0f;
                #pragma unroll
                for (int j = 0; j < 2; ++j) {
                    float v = acc[i][j][r] + bv[j];
                    v = v > 0.0f ? v : 0.0f;
                    s += v * rhv[j];
                }
                // reduce the 16 columns held by lanes sharing this C row
                #pragma unroll
                for (int off = 1; off < 16; off <<= 1)
                    s += __shfl_xor(s, off, 32);
                if (lm == 0) {
                    const int row = mBlock + wm + i * 16 + lh * 8 + r;
                    atomicAdd(&dots[(size_t)row * NHEAD + head], s);
                }
            }
        }
    } else {
        #pragma unroll
        for (int i = 0; i < 2; ++i) {
            #pragma unroll
            for (int j = 0; j < 2; ++j) {
                const int col = nBlock + wn + j * 16 + lm;
                #pragma unroll
                for (int r = 0; r < 8; ++r) {
                    const int row = mBlock + wm + i * 16 + lh * 8 + r;
                    float v = acc[i][j][r] + bv[j];
                    Out[(size_t)row * NN + col] = v > 0.0f ? v : 0.0f;
                }
            }
        }
    }
}

// ---------------------------------------------------------------------------
__global__ void zero_kernel(float* __restrict__ p, int n)
{
    int i = blockIdx.x * blockDim.x + threadIdx.x;
    if (i < n) p[i] = 0.0f;
}

// ---------------------------------------------------------------------------
// Sequential nearest-valid scans, one thread per (batch, head); thresholds
// the fused dots directly. Matches the reference lax.scan semantics exactly,
// including the backward map storing L-1-j when position j is valid.
// ---------------------------------------------------------------------------
__global__ void scan_kernel(const float* __restrict__ dots,
                            int* __restrict__ fa, int* __restrict__ fb,
                            int* __restrict__ ba, int* __restrict__ bb)
{
    int t = blockIdx.x * blockDim.x + threadIdx.x;
    if (t >= BSZ * NHEAD) return;
    int b = t / NHEAD, n = t % NHEAD;
    const size_t base = (size_t)b * LEN * NHEAD + n;
    const float* d = dots + base;
    int* pfa = fa + base;
    int* pfb = fb + base;
    int* pba = ba + base;
    int* pbb = bb + base;

    int a = 0, bp = 0;
    pfa[0] = 0; pfb[0] = 0;
    for (int j = 1; j < LEN; ++j) {
        if (d[(size_t)j * NHEAD] > 0.5f) { bp = a; a = j; }
        pfa[(size_t)j * NHEAD] = a;
        pfb[(size_t)j * NHEAD] = bp;
    }
    a = LEN - 1; bp = LEN - 1;
    pba[(size_t)(LEN - 1) * NHEAD] = LEN - 1;
    pbb[(size_t)(LEN - 1) * NHEAD] = LEN - 1;
    for (int j = LEN - 2; j >= 0; --j) {
        if (d[(size_t)j * NHEAD] > 0.5f) { bp = a; a = (LEN - 1) - j; }
        pba[(size_t)j * NHEAD] = a;
        pbb[(size_t)j * NHEAD] = bp;
    }
}

// ---------------------------------------------------------------------------
// context = sum_r w[n,r] * V1[b, map_r(l,n), n, :]   (float4 per thread;
// V1 is 64 MB and L2 is 192 MB, so the gathers largely hit L2)
// ---------------------------------------------------------------------------
__global__ void gather_kernel(const float* __restrict__ V1,
                              const int* __restrict__ fa, const int* __restrict__ fb,
                              const int* __restrict__ ba, const int* __restrict__ bb,
                              const float* __restrict__ Wb,   // (1,1,NH,1,4) flat
                              float* __restrict__ Out)
{
    int idx = blockIdx.x * blockDim.x + threadIdx.x;
    const int TOT = MM * NHEAD * (HD / 4);
    if (idx >= TOT) return;
    int h4  = idx % (HD / 4);
    int tmp = idx / (HD / 4);
    int n   = tmp % NHEAD;
    int r   = tmp / NHEAD;      // global row = b*LEN + l
    int b   = r / LEN;
    size_t mi = (size_t)r * NHEAD + n;
    int i0 = fa[mi], i1 = fb[mi], i2 = ba[mi], i3 = bb[mi];

    const size_t bbase = (size_t)b * LEN * NN;
    const int coff = n * HD + h4 * 4;
    f4 v0 = *(const f4*)(V1 + bbase + (size_t)i0 * NN + coff);
    f4 v1 = *(const f4*)(V1 + bbase + (size_t)i1 * NN + coff);
    f4 v2 = *(const f4*)(V1 + bbase + (size_t)i2 * NN + coff);
    f4 v3 = *(const f4*)(V1 + bbase + (size_t)i3 * NN + coff);

    float w0 = Wb[n * 4 + 0], w1 = Wb[n * 4 + 1];
    float w2 = Wb[n * 4 + 2], w3 = Wb[n * 4 + 3];
    f4 o = v0 * w0 + v1 * w1 + v2 * w2 + v3 * w3;
    *(f4*)(Out + (size_t)r * NN + coff) = o;
}

// ---------------------------------------------------------------------------
extern "C" void kernel_launch(void* const* d_in, const int* in_sizes, int n_in,
                              void* d_out, int out_size, void* d_ws, size_t ws_size,
                              hipStream_t stream)
{
    const float* X   = (const float*)d_in[0];  // hidden_states [8,2048,1024]
    const float* K1w = (const float*)d_in[1];  // [1024,1024]
    const float* K1b = (const float*)d_in[2];  // [1024]
    const float* V1w = (const float*)d_in[3];  // [1024,1024]
    const float* V1b = (const float*)d_in[4];  // [1024]
    const float* Wb  = (const float*)d_in[5];  // (1,1,16,1,4)
    const float* RH  = (const float*)d_in[6];  // [16,64]
    float* out = (float*)d_out;

    char* ws = (char*)d_ws;
    float* V1   = (float*)ws;                                   // 64 MB
    float* dots = (float*)(ws + (size_t)MM * NN * 4);           // 1 MB
    int*   fa   = (int*)(dots + (size_t)MM * NHEAD);
    int*   fb   = fa + (size_t)MM * NHEAD;
    int*   ba   = fb + (size_t)MM * NHEAD;
    int*   bb   = ba + (size_t)MM * NHEAD;

    const int ndots = MM * NHEAD;
    zero_kernel<<<(ndots + 255) / 256, 256, 0, stream>>>(dots, ndots);

    dim3 gGemm(MM / BM, NN / BN, 1);
    // K path: GEMM fused with dots reduction (K1 never hits HBM)
    gemm_fused_kernel<true><<<gGemm, 256, 0, stream>>>(X, K1w, K1b, RH, nullptr, dots);
    // V path: GEMM + bias + relu stored for the gather stage
    gemm_fused_kernel<false><<<gGemm, 256, 0, stream>>>(X, V1w, V1b, RH, V1, nullptr);

    scan_kernel<<<1, 128, 0, stream>>>(dots, fa, fb, ba, bb);

    const int tot = MM * NHEAD * (HD / 4);
    gather_kernel<<<(tot + 255) / 256, 256, 0, stream>>>(V1, fa, fb, ba, bb, Wb, out);
}